// SIGVAE_GIN_70257075028337
// MI455X (gfx1250) — compile-verified
//
#include <hip/hip_runtime.h>
#include <hip/hip_bf16.h>
#include <math.h>

typedef __attribute__((ext_vector_type(2))) float v2f;
typedef __attribute__((ext_vector_type(8))) float v8f;

#define WG 256

// ---------------------------------------------------------------------------
// zero a float buffer
__global__ __launch_bounds__(WG) void zero_kernel(float* __restrict__ p, int n) {
    int i = blockIdx.x * WG + threadIdx.x;
    if (i < n) p[i] = 0.0f;
}

// ---------------------------------------------------------------------------
// out[n, j] = sum_k concat(X[n,:F1], Y[n,:F2])[k] * W[k, j]
// W ((F1+F2) x OC) cached in LDS (max 128*32 = 4096 floats = 16KB)
__global__ __launch_bounds__(WG) void gemm_concat_kernel(
    const float* __restrict__ X, const float* __restrict__ Y,
    const float* __restrict__ W, float* __restrict__ out,
    int N, int F1, int F2, int OC) {
    __shared__ float sW[4096];
    const int K = F1 + F2;
    for (int i = threadIdx.x; i < K * OC; i += WG) sW[i] = W[i];
    __syncthreads();

    int tid = blockIdx.x * WG + threadIdx.x;
    if (tid >= N * OC) return;
    int n = tid / OC;
    int j = tid - n * OC;

    float acc = 0.0f;
    const float* xr = X + (size_t)n * F1;
    for (int k = 0; k < F1; ++k) acc += xr[k] * sW[k * OC + j];
    if (F2 > 0) {
        const float* yr = Y + (size_t)n * F2;
        for (int k = 0; k < F2; ++k) acc += yr[k] * sW[(F1 + k) * OC + j];
    }
    out[tid] = acc;
}

// ---------------------------------------------------------------------------
// scatter-add: a[dst[e], d] += t[src[e], d]   (one thread per (edge, dim))
__global__ __launch_bounds__(WG) void scatter_kernel(
    const float* __restrict__ t, const int* __restrict__ src,
    const int* __restrict__ dst, float* __restrict__ a, int E, int D) {
    int tid = blockIdx.x * WG + threadIdx.x;
    if (tid >= E * D) return;
    int e = tid / D;
    int d = tid - e * D;
    atomicAdd(&a[(size_t)dst[e] * D + d], t[(size_t)src[e] * D + d]);
}

// ---------------------------------------------------------------------------
// out = act((1+eps)*t + a + bias)
__global__ __launch_bounds__(WG) void combine_kernel(
    const float* __restrict__ t, const float* __restrict__ a,
    const float* __restrict__ bias, const float* __restrict__ eps, int eps_idx,
    float* __restrict__ out, int N, int D, int do_relu) {
    int tid = blockIdx.x * WG + threadIdx.x;
    if (tid >= N * D) return;
    int d = tid % D;
    float e = 1.0f + eps[eps_idx];
    float v = e * t[tid] + a[tid] + bias[d];
    if (do_relu) v = fmaxf(v, 0.0f);
    out[tid] = v;
}

// ---------------------------------------------------------------------------
// Z = reparam * exp(0.5*lg) + mu ; Zrk = Z * sigmoid(rk_logit[d])
__global__ __launch_bounds__(WG) void z_kernel(
    const float* __restrict__ mu, const float* __restrict__ lg,
    const float* __restrict__ reparam, const float* __restrict__ rk_logit,
    float* __restrict__ Z, float* __restrict__ Zrk, int N, int ZD) {
    int tid = blockIdx.x * WG + threadIdx.x;
    if (tid >= N * ZD) return;
    int d = tid % ZD;
    float sigma = expf(0.5f * lg[tid]);
    float z = reparam[tid] * sigma + mu[tid];
    float rk = 1.0f / (1.0f + expf(-rk_logit[d]));
    Z[tid] = z;
    Zrk[tid] = z * rk;
}

// ---------------------------------------------------------------------------
// Decoder: out[i,j] = 1 - exp(-exp(min( sum_d Zrk[i,d]*Z[j,d], 10 )))
// One wave computes one 16x16 tile via 4x V_WMMA_F32_16X16X4_F32 (K=16).
// A-frag (16x4 f32): M = lane%16; V0 holds K = kb+(lane<16?0:2), V1 = kb+(lane<16?1:3)
// B-frag (4x16 f32): N = lane%16; same K split (B row striped across lanes)
// C/D (16x16 f32, 8 VGPRs): row = v + (lane<16?0:8), col = lane%16
__global__ __launch_bounds__(WG) void decoder_wmma_kernel(
    const float* __restrict__ Zrk, const float* __restrict__ Z,
    float* __restrict__ out, int N) {
    int gtid = blockIdx.x * WG + threadIdx.x;
    int wave = gtid >> 5;
    int lane = threadIdx.x & 31;
    int tilesPerRow = N >> 4;                  // 512
    int ti = wave / tilesPerRow;
    int tj = wave - ti * tilesPerRow;
    if (ti >= tilesPerRow) return;             // exact fit: never taken

    int m = lane & 15;
    int hi = (lane >> 4) & 1;                  // 0 for lanes 0-15, 1 for 16-31

    const float* Arow = Zrk + (size_t)(ti * 16 + m) * 16;   // row M of A-tile
    const float* Brow = Z   + (size_t)(tj * 16 + m) * 16;   // col N of B-tile

    v8f c = {0.f, 0.f, 0.f, 0.f, 0.f, 0.f, 0.f, 0.f};
    #pragma unroll
    for (int kc = 0; kc < 4; ++kc) {
        int kb = kc * 4;
        v2f a, b;
        a.x = Arow[kb + (hi ? 2 : 0)];
        a.y = Arow[kb + (hi ? 3 : 1)];
        b.x = Brow[kb + (hi ? 2 : 0)];
        b.y = Brow[kb + (hi ? 3 : 1)];
        c = __builtin_amdgcn_wmma_f32_16x16x4_f32(
            /*neg_a=*/false, a, /*neg_b=*/false, b,
            /*c_mod=*/(short)0, c, /*reuse_a=*/false, /*reuse_b=*/false);
    }

    size_t colBase = (size_t)tj * 16 + m;
    int rowOff = hi ? 8 : 0;
    #pragma unroll
    for (int v = 0; v < 8; ++v) {
        float lg = c[v];
        lg = fminf(lg, 10.0f);
        float val = 1.0f - expf(-expf(lg));
        out[(size_t)(ti * 16 + rowOff + v) * (size_t)N + colBase] = val;
    }
}

// ---------------------------------------------------------------------------
extern "C" void kernel_launch(void* const* d_in, const int* in_sizes, int n_in,
                              void* d_out, int out_size, void* d_ws, size_t ws_size,
                              hipStream_t stream) {
    const int N = 8192, F = 64, ND = 64, H = 32, ZD = 16;
    const int E = in_sizes[1] / 2;

    const float* feat    = (const float*)d_in[0];
    const int*   src     = (const int*)d_in[1];            // edge_index[0]
    const int*   dst     = (const int*)d_in[1] + E;        // edge_index[1]
    const float* bern1   = (const float*)d_in[2] + (size_t)N * ND;  // sample 1 only
    const float* reparam = (const float*)d_in[3];          // [1,N,ZD]
    const float* u_W0  = (const float*)d_in[4];
    const float* u_b0  = (const float*)d_in[5];
    const float* u_W1  = (const float*)d_in[6];
    const float* u_b1  = (const float*)d_in[7];
    const float* u_eps = (const float*)d_in[8];
    const float* mu_W0  = (const float*)d_in[9];
    const float* mu_b0  = (const float*)d_in[10];
    const float* mu_W1  = (const float*)d_in[11];
    const float* mu_b1  = (const float*)d_in[12];
    const float* mu_eps = (const float*)d_in[13];
    const float* sg_W0  = (const float*)d_in[14];
    const float* sg_b0  = (const float*)d_in[15];
    const float* sg_W1  = (const float*)d_in[16];
    const float* sg_b1  = (const float*)d_in[17];
    const float* sg_eps = (const float*)d_in[18];
    const float* rk_logit = (const float*)d_in[19];

    // workspace layout (floats)
    const size_t NF = (size_t)N * H;   // 262144
    const size_t NZ = (size_t)N * ZD;  // 131072
    float* ws = (float*)d_ws;
    float* w_t   = ws;                 // [N,32] pre-agg GEMM result
    float* w_a   = ws + NF;            // [N,32] scatter accumulator
    float* w_h1  = ws + 2 * NF;        // GINu layer1 out
    float* w_h2  = ws + 3 * NF;        // GINu layer2 out
    float* w_m1  = ws + 4 * NF;        // mu/sg layer1 out
    float* w_t16 = ws + 5 * NF;        // [N,16] pre-agg
    float* w_a16 = ws + 5 * NF + NZ;   // [N,16] accumulator
    float* w_mu  = ws + 5 * NF + 2 * NZ;
    float* w_lg  = ws + 5 * NF + 3 * NZ;
    float* w_Z   = ws + 5 * NF + 4 * NZ;
    float* w_Zrk = ws + 5 * NF + 5 * NZ;

    const int gNF   = (int)(NF / WG);
    const int gNZ   = (int)(NZ / WG);
    const int gE32  = (E * 32 + WG - 1) / WG;
    const int gE16  = (E * 16 + WG - 1) / WG;

    // ---- GINu layer 0: concat(feat, bern1) [N,128] @ u_W0 -> 32, relu
    gemm_concat_kernel<<<gNF, WG, 0, stream>>>(feat, bern1, u_W0, w_t, N, F, ND, H);
    zero_kernel<<<gNF, WG, 0, stream>>>(w_a, (int)NF);
    scatter_kernel<<<gE32, WG, 0, stream>>>(w_t, src, dst, w_a, E, H);
    combine_kernel<<<gNF, WG, 0, stream>>>(w_t, w_a, u_b0, u_eps, 0, w_h1, N, H, 1);

    // ---- GINu layer 1: h1 [N,32] @ u_W1 -> 32, relu (outer relu is no-op)
    gemm_concat_kernel<<<gNF, WG, 0, stream>>>(w_h1, nullptr, u_W1, w_t, N, H, 0, H);
    zero_kernel<<<gNF, WG, 0, stream>>>(w_a, (int)NF);
    scatter_kernel<<<gE32, WG, 0, stream>>>(w_t, src, dst, w_a, E, H);
    combine_kernel<<<gNF, WG, 0, stream>>>(w_t, w_a, u_b1, u_eps, 1, w_h2, N, H, 1);

    // ---- GINmu: concat(feat, h2) [N,96] -> 32 (ident) -> 16 (ident)
    gemm_concat_kernel<<<gNF, WG, 0, stream>>>(feat, w_h2, mu_W0, w_t, N, F, H, H);
    zero_kernel<<<gNF, WG, 0, stream>>>(w_a, (int)NF);
    scatter_kernel<<<gE32, WG, 0, stream>>>(w_t, src, dst, w_a, E, H);
    combine_kernel<<<gNF, WG, 0, stream>>>(w_t, w_a, mu_b0, mu_eps, 0, w_m1, N, H, 0);

    gemm_concat_kernel<<<gNZ, WG, 0, stream>>>(w_m1, nullptr, mu_W1, w_t16, N, H, 0, ZD);
    zero_kernel<<<gNZ, WG, 0, stream>>>(w_a16, (int)NZ);
    scatter_kernel<<<gE16, WG, 0, stream>>>(w_t16, src, dst, w_a16, E, ZD);
    combine_kernel<<<gNZ, WG, 0, stream>>>(w_t16, w_a16, mu_b1, mu_eps, 1, w_mu, N, ZD, 0);

    // ---- GINsigma: same structure
    gemm_concat_kernel<<<gNF, WG, 0, stream>>>(feat, w_h2, sg_W0, w_t, N, F, H, H);
    zero_kernel<<<gNF, WG, 0, stream>>>(w_a, (int)NF);
    scatter_kernel<<<gE32, WG, 0, stream>>>(w_t, src, dst, w_a, E, H);
    combine_kernel<<<gNF, WG, 0, stream>>>(w_t, w_a, sg_b0, sg_eps, 0, w_m1, N, H, 0);

    gemm_concat_kernel<<<gNZ, WG, 0, stream>>>(w_m1, nullptr, sg_W1, w_t16, N, H, 0, ZD);
    zero_kernel<<<gNZ, WG, 0, stream>>>(w_a16, (int)NZ);
    scatter_kernel<<<gE16, WG, 0, stream>>>(w_t16, src, dst, w_a16, E, ZD);
    combine_kernel<<<gNZ, WG, 0, stream>>>(w_t16, w_a16, sg_b1, sg_eps, 1, w_lg, N, ZD, 0);

    // ---- reparameterization + rk scaling
    z_kernel<<<gNZ, WG, 0, stream>>>(w_mu, w_lg, reparam, rk_logit, w_Z, w_Zrk, N, ZD);

    // ---- decoder: (Z∘rk) @ Z^T via f32 WMMA, fused clamp + 1-exp(-exp(.))
    // 512x512 tiles, 1 wave (32 lanes) per 16x16 tile -> 262144 waves
    int tiles = (N / 16) * (N / 16);
    int dec_blocks = tiles * 32 / WG;   // 32768
    decoder_wmma_kernel<<<dec_blocks, WG, 0, stream>>>(w_Zrk, w_Z, (float*)d_out, N);
}